// HeteroGraphSAGE_19104014532708
// MI455X (gfx1250) — compile-verified
//
#include <hip/hip_runtime.h>
#include <hip/hip_bf16.h>

#define C_DIM 128
#define SW_STRIDE 136   // LDS row stride in floats (bank-conflict-free across lane halves)

typedef float v2f __attribute__((ext_vector_type(2)));
typedef float v8f __attribute__((ext_vector_type(8)));

// ---------------------------------------------------------------- zero buffer
__global__ void hg_zero(float* __restrict__ p, long n) {
  long i = ((long)blockIdx.x * blockDim.x + threadIdx.x) * 4;
  if (i + 3 < n) {
    float4 z = make_float4(0.f, 0.f, 0.f, 0.f);
    *reinterpret_cast<float4*>(p + i) = z;
  } else {
    for (int k = 0; k < 4; ++k)
      if (i + k < n) p[i + k] = 0.f;
  }
}

// ------------------------------------------------- edge scatter (segment sum)
// one wave (32 lanes) per edge; each lane handles 4 contiguous features
__global__ void hg_scatter(const float* __restrict__ xsrc, const int* __restrict__ src,
                           const int* __restrict__ dst, float* __restrict__ agg,
                           float* __restrict__ cnt, int E) {
  int gid = blockIdx.x * blockDim.x + threadIdx.x;
  int e = gid >> 5;
  if (e >= E) return;
  int lane = threadIdx.x & 31;
  int s = src[e];
  int d = dst[e];
  const float4 v =
      *reinterpret_cast<const float4*>(xsrc + (size_t)s * C_DIM + lane * 4);
  float* base = agg + (size_t)d * C_DIM + lane * 4;
  atomicAdd(base + 0, v.x);
  atomicAdd(base + 1, v.y);
  atomicAdd(base + 2, v.z);
  atomicAdd(base + 3, v.w);
  if (lane == 0) atomicAdd(cnt + d, 1.0f);
}

// --------------------------------------------------------- WMMA fp32 GEMM
// out[M,128] (+)= (rowscale ? diag(1/max(cnt,1)) : I) * A[M,128] @ W[128,128] (+ bias)
// one wave computes a 16x128 strip via 8 accumulator tiles of v_wmma_f32_16x16x4_f32
__global__ __launch_bounds__(256) void hg_gemm_wmma(
    const float* __restrict__ A, const float* __restrict__ cnt,
    const float* __restrict__ W, const float* __restrict__ bias,
    float* __restrict__ out, int M, int beta) {
  __shared__ float sW[64 * SW_STRIDE];  // 34.8 KB: half of W per stage

  const int tid = threadIdx.x;
  const int wave = tid >> 5;
  const int lane = tid & 31;
  const int half = lane >> 4;   // lane half selects K (A/B) or M (C/D) split
  const int l16 = lane & 15;
  const int mbase = (blockIdx.x * 8 + wave) * 16;

  const int arow = mbase + l16;
  const int arow_c = (arow < M) ? arow : (M - 1);  // clamp: OOB rows never stored
  float s = 1.0f;
  if (cnt) s = 1.0f / fmaxf(cnt[arow_c], 1.0f);

  v8f acc[8];
#pragma unroll
  for (int nt = 0; nt < 8; ++nt) {
#pragma unroll
    for (int r = 0; r < 8; ++r) {
      const int row = mbase + r + 8 * half;   // C/D layout: VGPR r -> row r / 8+r
      const int col = nt * 16 + l16;
      float v = 0.f;
      if (beta && row < M) v = out[(size_t)row * C_DIM + col];
      if (bias) v += bias[col];
      acc[nt][r] = v;
    }
  }

  const float* Arow = A + (size_t)arow_c * C_DIM;

  for (int stage = 0; stage < 2; ++stage) {
    __syncthreads();
    for (int i = tid; i < 64 * C_DIM; i += 256) {
      const int r = i >> 7;
      const int c = i & 127;
      sW[r * SW_STRIDE + c] = W[(size_t)(stage * 64 + r) * C_DIM + c];
    }
    __syncthreads();
#pragma unroll
    for (int kk = 0; kk < 16; ++kk) {
      const int klocal = kk * 4 + half * 2;     // A layout: lanes0-15 K={0,1}, 16-31 K={2,3}
      const int kglob = stage * 64 + klocal;
      v2f a = *reinterpret_cast<const v2f*>(Arow + kglob);
      a = a * s;                                 // fold mean-divide into A
#pragma unroll
      for (int nt = 0; nt < 8; ++nt) {
        v2f b;
        b.x = sW[klocal * SW_STRIDE + nt * 16 + l16];
        b.y = sW[(klocal + 1) * SW_STRIDE + nt * 16 + l16];
        acc[nt] = __builtin_amdgcn_wmma_f32_16x16x4_f32(
            false, a, false, b, (short)0, acc[nt], false, false);
      }
    }
  }

#pragma unroll
  for (int nt = 0; nt < 8; ++nt) {
#pragma unroll
    for (int r = 0; r < 8; ++r) {
      const int row = mbase + r + 8 * half;
      if (row < M) out[(size_t)row * C_DIM + nt * 16 + l16] = acc[nt][r];
    }
  }
}

// ------------------------------------------------------- LayerNorm + ReLU
// one wave per row of 128; wave32 shuffle reductions
__global__ void hg_ln_relu(float* __restrict__ x, const float* __restrict__ w,
                           const float* __restrict__ b, int M) {
  int gid = blockIdx.x * blockDim.x + threadIdx.x;
  int row = gid >> 5;
  if (row >= M) return;
  int lane = threadIdx.x & 31;
  float* rp = x + (size_t)row * C_DIM + lane * 4;
  float4 v = *reinterpret_cast<float4*>(rp);
  float sum = v.x + v.y + v.z + v.w;
#pragma unroll
  for (int off = 16; off >= 1; off >>= 1) sum += __shfl_xor(sum, off, 32);
  const float mu = sum * (1.0f / C_DIM);
  const float d0 = v.x - mu, d1 = v.y - mu, d2 = v.z - mu, d3 = v.w - mu;
  float ss = d0 * d0 + d1 * d1 + d2 * d2 + d3 * d3;
#pragma unroll
  for (int off = 16; off >= 1; off >>= 1) ss += __shfl_xor(ss, off, 32);
  const float rstd = rsqrtf(ss * (1.0f / C_DIM) + 1e-5f);
  const float4 wv = *reinterpret_cast<const float4*>(w + lane * 4);
  const float4 bv = *reinterpret_cast<const float4*>(b + lane * 4);
  float4 o;
  o.x = fmaxf(d0 * rstd * wv.x + bv.x, 0.f);
  o.y = fmaxf(d1 * rstd * wv.y + bv.y, 0.f);
  o.z = fmaxf(d2 * rstd * wv.z + bv.z, 0.f);
  o.w = fmaxf(d3 * rstd * wv.w + bv.w, 0.f);
  *reinterpret_cast<float4*>(rp) = o;
}

// ---------------------------------------------------------------- launcher
extern "C" void kernel_launch(void* const* d_in, const int* in_sizes, int n_in,
                              void* d_out, int out_size, void* d_ws, size_t ws_size,
                              hipStream_t stream) {
  const float* x_user = (const float*)d_in[0];
  const float* x_item = (const float*)d_in[1];
  const int* ei_ui = (const int*)d_in[2];
  const int* ei_iu = (const int*)d_in[3];
  const int* ei_uu = (const int*)d_in[4];
  const float* Wl = (const float*)d_in[5];
  const float* bl = (const float*)d_in[6];
  const float* Wr = (const float*)d_in[7];
  const float* lnw = (const float*)d_in[8];
  const float* lnb = (const float*)d_in[9];

  const int C = C_DIM;
  const int NU = in_sizes[0] / C;
  const int NI = in_sizes[1] / C;
  const int E = in_sizes[2] / 2;
  const int NMAX = (NU > NI) ? NU : NI;
  const int L = 2;

  float* xu_buf = (float*)d_ws;
  float* xi_buf = xu_buf + (size_t)NU * C;
  float* agg = xi_buf + (size_t)NI * C;
  float* cnt = agg + (size_t)NMAX * C;
  float* outp = (float*)d_out;

  auto zero = [&](float* p, long n) {
    long thr = (n + 3) / 4;
    int blocks = (int)((thr + 255) / 256);
    hg_zero<<<blocks, 256, 0, stream>>>(p, n);
  };
  auto scatter = [&](const float* xs, const int* ei) {
    long thr = (long)E * 32;
    int blocks = (int)((thr + 255) / 256);
    hg_scatter<<<blocks, 256, 0, stream>>>(xs, ei, ei + E, agg, cnt, E);
  };
  auto gemm = [&](const float* A, const float* cv, const float* W,
                  const float* bias, float* out, int M, int beta) {
    int blocks = (M + 127) / 128;
    hg_gemm_wmma<<<blocks, 256, 0, stream>>>(A, cv, W, bias, out, M, beta);
  };
  auto lnrelu = [&](float* x, const float* w, const float* b, int M) {
    long thr = (long)M * 32;
    int blocks = (int)((thr + 255) / 256);
    hg_ln_relu<<<blocks, 256, 0, stream>>>(x, w, b, M);
  };

  for (int l = 0; l < L; ++l) {
    const float* xu_src = (l == 0) ? x_user : xu_buf;
    const float* xi_src = (l == 0) ? x_item : xi_buf;
    float* xu_dst = (l == L - 1) ? outp : xu_buf;
    float* xi_dst = (l == L - 1) ? (outp + (size_t)NU * C) : xi_buf;

    const size_t w0 = ((size_t)l * 3 + 0) * C * C;
    const size_t w1 = ((size_t)l * 3 + 1) * C * C;
    const size_t w2 = ((size_t)l * 3 + 2) * C * C;
    const size_t b0 = ((size_t)l * 3 + 0) * C;
    const size_t b1 = ((size_t)l * 3 + 1) * C;
    const size_t b2 = ((size_t)l * 3 + 2) * C;

    // ---- edge type 0: user -> item ----
    zero(agg, (long)NI * C);
    zero(cnt, NI);
    scatter(xu_src, ei_ui);
    gemm(agg, cnt, Wl + w0, bl + b0, xi_dst, NI, /*beta=*/0);
    gemm(xi_src, nullptr, Wr + w0, nullptr, xi_dst, NI, /*beta=*/1);

    // ---- edge type 1: item -> user ----
    zero(agg, (long)NU * C);
    zero(cnt, NU);
    scatter(xi_src, ei_iu);
    gemm(agg, cnt, Wl + w1, bl + b1, xu_dst, NU, /*beta=*/0);
    gemm(xu_src, nullptr, Wr + w1, nullptr, xu_dst, NU, /*beta=*/1);

    // ---- edge type 2: user -> user (HeteroConv sum) ----
    zero(agg, (long)NU * C);
    zero(cnt, NU);
    scatter(xu_src, ei_uu);
    gemm(agg, cnt, Wl + w2, bl + b2, xu_dst, NU, /*beta=*/1);
    gemm(xu_src, nullptr, Wr + w2, nullptr, xu_dst, NU, /*beta=*/1);

    // ---- per-node-type LayerNorm + ReLU (in place) ----
    lnrelu(xu_dst, lnw + ((size_t)l * 2 + 0) * C, lnb + ((size_t)l * 2 + 0) * C, NU);
    lnrelu(xi_dst, lnw + ((size_t)l * 2 + 1) * C, lnb + ((size_t)l * 2 + 1) * C, NI);
  }
}